// GINetConvLayer_4836133175445
// MI455X (gfx1250) — compile-verified
//
#include <hip/hip_runtime.h>

typedef __attribute__((ext_vector_type(2))) float v2f;
typedef __attribute__((ext_vector_type(8))) float v8f;

// ---------------------------------------------------------------------------
// Exact simplification of the reference:
//   softmax over axis=1 of an [E,1] tensor is identically 1.0  =>  alpha == 1
//   z = segment_sum(x[col] @ W_fc^T, row)
//     = segment_sum(x[col], row) @ W_fc^T        (segment_sum is linear)
// So edge_attr / W_edge / W_att / xrow are dead, and the GEMM shrinks from
// E=800k rows to N=50k rows (16x fewer FLOPs).
//
// Roofline on MI455X (23.3 TB/s HBM, 192 MB L2):
//   phase 1 (gather + scatter-add): ~205 MB gathered reads (x = 12.8 MB, L2
//            resident) + ~205 MB of global_atomic_add_f32  -> dominant cost
//   phase 2 (g @ W^T): 410 MFLOP, ~26 MB traffic           -> negligible;
//            done with V_WMMA_F32_16X16X4_F32 to keep exact fp32 semantics
//            while using the CDNA5 matrix pipe.
// ---------------------------------------------------------------------------

__global__ void zero_f4_kernel(float4* __restrict__ p, int n4) {
  int i = blockIdx.x * blockDim.x + threadIdx.x;
  if (i < n4) p[i] = make_float4(0.f, 0.f, 0.f, 0.f);
}

// One thread per (edge, 4-channel group): 16 consecutive threads cooperatively
// gather one 256B row of x (coalesced float4 loads) and scatter-add it.
__global__ void edge_aggregate_kernel(const float* __restrict__ x,
                                      const int* __restrict__ ei,   // [2,E]
                                      float* __restrict__ g,
                                      int E) {
  long long t = (long long)blockIdx.x * blockDim.x + threadIdx.x;
  long long e = t >> 4;
  if (e >= E) return;
  const int c4 = (int)(t & 15) << 2;
  const long long dst = ei[e];        // row (scatter destination)
  const long long src = ei[E + e];    // col (gather source)
  const float4 v = *reinterpret_cast<const float4*>(x + src * 64 + c4);
  float* out = g + dst * 64 + c4;
  unsafeAtomicAdd(out + 0, v.x);      // global_atomic_add_f32 (no CAS loop)
  unsafeAtomicAdd(out + 1, v.y);
  unsafeAtomicAdd(out + 2, v.z);
  unsafeAtomicAdd(out + 3, v.w);
}

// z[N,64] = g[N,64] @ W^T,  W is [64,64] row-major (W[n*64+k]).
// Block = 128 threads = 4 waves; each block owns a 16-row tile, wave w owns
// output columns [16w, 16w+16). K=64 swept as 16 x V_WMMA_F32_16X16X4_F32.
__global__ void __launch_bounds__(128)
gemm_wmma_f32_kernel(const float* __restrict__ g,
                     const float* __restrict__ W,
                     float* __restrict__ z,
                     int rowTiles) {
  __shared__ float ash[16 * 68];            // 16x64 A tile, padded (68) to kill
                                            // LDS bank conflicts on ds_load_b64
  const int lane = threadIdx.x & 31;
  const int wave = threadIdx.x >> 5;        // 0..3 -> 16-wide column tile
  const int half = lane >> 4;               // 0/1 half-wave (ISA A/D layout)
  const int l16  = lane & 15;
  const int c0   = wave << 4;

  // Preload all B fragments for the K sweep: B[k][n] = W[n][k].
  // 16 steps x v2f = 32 VGPRs/lane; W is 16 KB -> L2/WGP$ resident.
  v2f bfrag[16];
#pragma unroll
  for (int ks = 0; ks < 16; ++ks) {
    const int k0 = ks * 4 + half * 2;
    const float* wp = W + (c0 + l16) * 64 + k0;
    bfrag[ks].x = wp[0];
    bfrag[ks].y = wp[1];
  }

  for (int tile = blockIdx.x; tile < rowTiles; tile += gridDim.x) {
    const long long rowBase = (long long)tile << 4;

    // Stage the 16x64 fp32 A tile into LDS (coalesced float4 global loads).
    const float* src = g + rowBase * 64;
    for (int idx = threadIdx.x; idx < 256; idx += 128) {
      const int r  = idx >> 4;
      const int c4 = (idx & 15) << 2;
      *reinterpret_cast<float4*>(&ash[r * 68 + c4]) =
          *reinterpret_cast<const float4*>(&src[r * 64 + c4]);
    }
    __syncthreads();

    v8f acc = {};
#pragma unroll
    for (int ks = 0; ks < 16; ++ks) {
      const int k0 = ks * 4 + half * 2;     // A layout: lanes 0-15 K={k0,k0+1},
      const float* ap = &ash[l16 * 68 + k0];//           lanes 16-31 K={k0+2,k0+3}
      v2f afrag;
      afrag.x = ap[0];
      afrag.y = ap[1];
      acc = __builtin_amdgcn_wmma_f32_16x16x4_f32(
          /*neg_a=*/false, afrag, /*neg_b=*/false, bfrag[ks],
          /*c_mod=*/(short)0, acc, /*reuse_a=*/false, /*reuse_b=*/false);
    }

    // D layout: VGPR j -> row (j + 8*half), col (c0 + l16).
#pragma unroll
    for (int j = 0; j < 8; ++j) {
      z[(rowBase + j + 8 * half) * 64 + c0 + l16] = acc[j];
    }
    __syncthreads();
  }
}

// Scalar tail for N % 16 rows (not taken for N=50000, kept for generality).
__global__ void gemm_tail_kernel(const float* __restrict__ g,
                                 const float* __restrict__ W,
                                 float* __restrict__ z,
                                 int rowStart, int N) {
  int t = blockIdx.x * blockDim.x + threadIdx.x;
  int row = rowStart + (t >> 6);
  int col = t & 63;
  if (row >= N) return;
  float s = 0.f;
#pragma unroll
  for (int k = 0; k < 64; ++k) s += g[(long long)row * 64 + k] * W[col * 64 + k];
  z[(long long)row * 64 + col] = s;
}

extern "C" void kernel_launch(void* const* d_in, const int* in_sizes, int n_in,
                              void* d_out, int out_size, void* d_ws, size_t ws_size,
                              hipStream_t stream) {
  const float* x   = (const float*)d_in[0];   // [N,64]
  const int*   ei  = (const int*)d_in[1];     // [2,E]
  const float* Wfc = (const float*)d_in[3];   // [64,64]
  // d_in[2] (edge_attr), d_in[4] (W_edge), d_in[5] (W_att) are mathematically
  // dead: softmax over a singleton axis is identically 1.

  float* z = (float*)d_out;
  float* g = (float*)d_ws;                    // [N,64] fp32 aggregation scratch

  const int C = 64;
  const int N = in_sizes[0] / C;
  const int E = in_sizes[1] / 2;

  // 1) zero the aggregation buffer (d_ws is poisoned, must re-zero every call)
  {
    const int n4 = N * (C / 4);
    zero_f4_kernel<<<(n4 + 255) / 256, 256, 0, stream>>>((float4*)g, n4);
  }

  // 2) g[row[e]] += x[col[e]]  (dominant, memory/atomic bound)
  {
    const long long T = (long long)E * 16;
    const int blocks = (int)((T + 255) / 256);
    edge_aggregate_kernel<<<blocks, 256, 0, stream>>>(x, ei, g, E);
  }

  // 3) z = g @ Wfc^T on the WMMA pipe (exact fp32)
  {
    const int rowTiles = N / 16;
    if (rowTiles > 0) {
      const int blocks = rowTiles < 2048 ? rowTiles : 2048;
      gemm_wmma_f32_kernel<<<blocks, 128, 0, stream>>>(g, Wfc, z, rowTiles);
    }
    const int rem = N & 15;
    if (rem > 0) {
      const int t = rem * C;
      gemm_tail_kernel<<<(t + 63) / 64, 64, 0, stream>>>(g, Wfc, z, N - rem, N);
    }
  }
}